// cross_attention_lstm_changed_83141976916070
// MI455X (gfx1250) — compile-verified
//
#include <hip/hip_runtime.h>
#include <math.h>
#include <stdint.h>

// ---------------------------------------------------------------------------
// CDNA5 (gfx1250) fp32 WMMA 16x16x4: A/B = 2 VGPRs, C/D = 8 VGPRs (wave32).
// A layout (16x4 MxK): lanes 0-15 -> row M=lane, regs hold K=0,1;
//                      lanes 16-31 -> row M=lane-16, regs hold K=2,3.
// C/D: reg r, lane l -> row r + 8*(l>=16), col l&15.
// ---------------------------------------------------------------------------
typedef __attribute__((ext_vector_type(2))) float f32x2;
typedef __attribute__((ext_vector_type(8))) float f32x8;

#define HB 512
#define BB 64
#define VV 10000
#define TT 64
#define APAD 4   // LDS row pad (floats): row stride K+4 -> bank 4r per row, conflict-free

__device__ __forceinline__ float sigmoidf_(float x) { return 1.0f / (1.0f + expf(-x)); }

// Async global->LDS copy, 16 bytes per lane (ASYNCcnt-tracked).  lds_off is the
// wave-relative LDS byte offset (low 32 bits of a __shared__ generic pointer).
__device__ __forceinline__ void async_copy_b128(uint32_t lds_off, const float* gaddr) {
    asm volatile("global_load_async_to_lds_b128 %0, %1, off"
                 :: "v"(lds_off), "v"(gaddr)
                 : "memory");
}
__device__ __forceinline__ void wait_async0() {
    asm volatile("s_wait_asynccnt 0x0" ::: "memory");
}

__device__ __forceinline__ f32x8 wmma4(f32x2 a, f32x2 b, f32x8 acc) {
    return __builtin_amdgcn_wmma_f32_16x16x4_f32(false, a, false, b, (short)0, acc,
                                                 false, false);
}

// ---------------------------------------------------------------------------
// Fused encoder: per 16-row tile of the flattened [B*512, 49] image, compute
// ci rows = tanh(img @ W_e2d^T + b) into LDS, then WMMA-GEMM the LDS tile
// against W_k / W_v (K=512) producing keys/values [B,512,512].
// block = (32, 8); grid = B*512/16 = 2048.
// ---------------------------------------------------------------------------
__global__ __launch_bounds__(256)
void encoder_kv_kernel(const float* __restrict__ img,
                       const float* __restrict__ W_e2d, const float* __restrict__ b_e2d,
                       const float* __restrict__ W_k, const float* __restrict__ b_k,
                       const float* __restrict__ W_v, const float* __restrict__ b_v,
                       float* __restrict__ keys, float* __restrict__ values) {
    __shared__ float ci_s[16][HB + APAD];
    const int m0 = blockIdx.x * 16;
    const int tid = threadIdx.y * 32 + threadIdx.x;

    for (int idx = tid; idx < 16 * HB; idx += 256) {
        const int r = idx >> 9, hh = idx & (HB - 1);
        const float* irow = img + (long long)(m0 + r) * 49;
        const float* wrow = W_e2d + hh * 49;
        float s = b_e2d[hh];
        for (int t = 0; t < 49; ++t) s += irow[t] * wrow[t];
        ci_s[r][hh] = tanhf(s);
    }
    __syncthreads();

    const int lane = threadIdx.x, l15 = lane & 15, hi = lane >> 4;
    for (int tt = threadIdx.y; tt < 64; tt += 8) {
        const int mat = tt >> 5, nt = tt & 31;
        const float* W    = mat ? W_v : W_k;
        const float* bias = mat ? b_v : b_k;
        float* out        = mat ? values : keys;
        const float* ap = &ci_s[l15][2 * hi];
        const float* bp = W + (long long)(nt * 16 + l15) * HB + 2 * hi;
        f32x8 acc = {};
        for (int k = 0; k < HB; k += 4) {
            f32x2 a; a.x = ap[k]; a.y = ap[k + 1];
            f32x2 b; b.x = bp[k]; b.y = bp[k + 1];
            acc = wmma4(a, b, acc);
        }
        const float bb = bias[nt * 16 + l15];
        for (int r = 0; r < 8; ++r) {
            const int row = m0 + r + 8 * hi;
            out[(long long)row * HB + nt * 16 + l15] = tanhf(acc[r] + bb);
        }
    }
}

// ---------------------------------------------------------------------------
// M-stripe WMMA GEMM, M = MT*16 rows (A contiguous [M,K]):
//   C = act(A1@B1^T + [A2@B2^T] + bias1 + bias2)
// A sources are async-staged into padded LDS once per block; each wave owns a
// 16-column stripe over ALL M (MT accumulators), so each B fragment load feeds
// MT WMMAs and the weight matrix is streamed exactly once per launch.
// Optional second strided output C2 (scatters logits into d_out [B,V,T]).
// block = (32, 8); dynamic LDS = MT*16*(K1+4) [+ MT*16*(K2+4)] floats.
// ---------------------------------------------------------------------------
template <int MT>
__global__ __launch_bounds__(256)
void gemm_stripe_kernel(const float* __restrict__ A1, int K1,
                        const float* __restrict__ B1, int ldb1,
                        const float* __restrict__ A2, int K2,
                        const float* __restrict__ B2, int ldb2,
                        const float* __restrict__ bias1, const float* __restrict__ bias2,
                        float* __restrict__ C, long long ldc,
                        float* __restrict__ C2, long long c2_rs, long long c2_cs,
                        int N, int act) {
    extern __shared__ float shA[];
    const int tid = threadIdx.y * 32 + threadIdx.x;
    const uint32_t lds0 = (uint32_t)(uintptr_t)(void*)shA;   // LDS byte offset
    const int ld1 = K1 + APAD;
    const int a2base = MT * 16 * ld1;                        // floats

    // Async-stage A1 (and A2) into LDS with per-row padding.
    for (int i = tid * 4; i < MT * 16 * K1; i += 1024) {
        const int r = i / K1, c = i - r * K1;
        async_copy_b128(lds0 + (uint32_t)(r * ld1 + c) * 4u, A1 + i);
    }
    if (A2) {
        const int ld2 = K2 + APAD;
        for (int i = tid * 4; i < MT * 16 * K2; i += 1024) {
            const int r = i / K2, c = i - r * K2;
            async_copy_b128(lds0 + (uint32_t)(a2base + r * ld2 + c) * 4u, A2 + i);
        }
    }
    wait_async0();
    __syncthreads();

    const int nt = blockIdx.x * blockDim.y + threadIdx.y;    // wave-uniform
    if (nt >= (N >> 4)) return;
    const int lane = threadIdx.x, l15 = lane & 15, hi = lane >> 4;

    f32x8 acc[MT];
#pragma unroll
    for (int m = 0; m < MT; ++m) acc[m] = (f32x8){};

    {
        const float* bp = B1 + (long long)(nt * 16 + l15) * ldb1 + 2 * hi;
        const float* ap = shA + l15 * ld1 + 2 * hi;
        for (int k = 0; k < K1; k += 4) {
            f32x2 b; b.x = bp[k]; b.y = bp[k + 1];
#pragma unroll
            for (int m = 0; m < MT; ++m) {
                f32x2 a; a.x = ap[m * 16 * ld1 + k]; a.y = ap[m * 16 * ld1 + k + 1];
                acc[m] = wmma4(a, b, acc[m]);
            }
        }
    }
    if (A2) {
        const int ld2 = K2 + APAD;
        const float* bp = B2 + (long long)(nt * 16 + l15) * ldb2 + 2 * hi;
        const float* ap = shA + a2base + l15 * ld2 + 2 * hi;
        for (int k = 0; k < K2; k += 4) {
            f32x2 b; b.x = bp[k]; b.y = bp[k + 1];
#pragma unroll
            for (int m = 0; m < MT; ++m) {
                f32x2 a; a.x = ap[m * 16 * ld2 + k]; a.y = ap[m * 16 * ld2 + k + 1];
                acc[m] = wmma4(a, b, acc[m]);
            }
        }
    }

    const int col = nt * 16 + l15;
    float bsum = 0.0f;
    if (bias1) bsum += bias1[col];
    if (bias2) bsum += bias2[col];
#pragma unroll
    for (int m = 0; m < MT; ++m) {
        for (int r = 0; r < 8; ++r) {
            const long long row = m * 16 + r + 8 * hi;
            float v = acc[m][r] + bsum;
            if (act) v = tanhf(v);
            C[row * ldc + col] = v;
            if (C2) C2[row * c2_rs + (long long)col * c2_cs] = v;
        }
    }
}

// ---------------------------------------------------------------------------
// Attention core for one (state, batch) pair: scores = keys[b] @ q, softmax
// over 512 channels, ctx = weights @ values[b].
// grid = (64, 2): y=0 -> h-attend rows, y=1 -> c-attend rows (q/ctx [2,64,512]).
// ---------------------------------------------------------------------------
__global__ __launch_bounds__(256)
void attn_kernel(const float* __restrict__ q, const float* __restrict__ keys,
                 const float* __restrict__ values, float* __restrict__ ctx) {
    const int b = blockIdx.x;
    const long long qrow = (long long)(blockIdx.y * BB + b) * HB;
    __shared__ float w_s[HB];
    __shared__ float red[256];
    const int tid = threadIdx.x;
    const int wave = tid >> 5, lane = tid & 31;
    const float* qb = q + qrow;
    const float* kb = keys + (long long)b * HB * HB;

    for (int c = wave; c < HB; c += 8) {
        const float* kr = kb + (long long)c * HB;
        float s = 0.0f;
        for (int k = lane; k < HB; k += 32) s += kr[k] * qb[k];
        for (int off = 16; off; off >>= 1) s += __shfl_down(s, off, 32);
        if (lane == 0) w_s[c] = s;
    }
    __syncthreads();

    float m = -INFINITY;
    for (int c = tid; c < HB; c += 256) m = fmaxf(m, w_s[c]);
    red[tid] = m; __syncthreads();
    for (int s = 128; s; s >>= 1) { if (tid < s) red[tid] = fmaxf(red[tid], red[tid + s]); __syncthreads(); }
    const float mx = red[0]; __syncthreads();
    float sum = 0.0f;
    for (int c = tid; c < HB; c += 256) { float e = expf(w_s[c] - mx); w_s[c] = e; sum += e; }
    red[tid] = sum; __syncthreads();
    for (int s = 128; s; s >>= 1) { if (tid < s) red[tid] += red[tid + s]; __syncthreads(); }
    const float inv = 1.0f / red[0]; __syncthreads();
    for (int c = tid; c < HB; c += 256) w_s[c] *= inv;
    __syncthreads();

    const float* vb = values + (long long)b * HB * HB;
    for (int d = tid; d < HB; d += 256) {
        float acc = 0.0f;
        for (int c = 0; c < HB; ++c) acc += w_s[c] * vb[(long long)c * HB + d];
        ctx[qrow + d] = acc;
    }
}

// ---------------------------------------------------------------------------
__global__ __launch_bounds__(256)
void lstm_update_kernel(const float* __restrict__ gates, const float* __restrict__ c_in,
                        float* __restrict__ hL, float* __restrict__ cL) {
    const int idx = blockIdx.x * 256 + threadIdx.x;
    if (idx >= BB * HB) return;
    const int b = idx >> 9, j = idx & (HB - 1);
    const float* g = gates + (long long)b * 4 * HB;
    const float i  = sigmoidf_(g[j]);
    const float f  = sigmoidf_(g[HB + j]);
    const float gg = tanhf(g[2 * HB + j]);
    const float o  = sigmoidf_(g[3 * HB + j]);
    const float c  = f * c_in[idx] + i * gg;
    cL[idx] = c;
    hL[idx] = o * tanhf(c);
}

// First-max argmax over logits[b, 0:10000] (matches jnp.argmax), gather embed.
__global__ __launch_bounds__(256)
void argmax_gather_kernel(const float* __restrict__ logits, const float* __restrict__ embed,
                          float* __restrict__ x) {
    const int b = blockIdx.x;
    __shared__ float bv[256];
    __shared__ int   bi[256];
    const int tid = threadIdx.x;
    float best = -INFINITY; int bidx = 0x7fffffff;
    const float* lb = logits + (long long)b * VV;
    for (int v = tid; v < VV; v += 256) {
        const float val = lb[v];
        if (val > best || (val == best && v < bidx)) { best = val; bidx = v; }
    }
    bv[tid] = best; bi[tid] = bidx; __syncthreads();
    for (int s = 128; s; s >>= 1) {
        if (tid < s) {
            if (bv[tid + s] > bv[tid] || (bv[tid + s] == bv[tid] && bi[tid + s] < bi[tid])) {
                bv[tid] = bv[tid + s]; bi[tid] = bi[tid + s];
            }
        }
        __syncthreads();
    }
    const int token = bi[0];
    const float* erow = embed + (long long)token * HB;
    for (int i = tid; i < HB; i += 256) x[(long long)b * HB + i] = erow[i];
}

__global__ __launch_bounds__(256)
void init_state_kernel(const float* __restrict__ pooled, const float* __restrict__ embed,
                       float* __restrict__ h, float* __restrict__ c, float* __restrict__ x) {
    const int idx = blockIdx.x * 256 + threadIdx.x;
    if (idx >= BB * HB) return;
    const float p = pooled[idx];
    h[idx] = p; c[idx] = p;
    x[idx] = embed[idx & (HB - 1)];
}

// res[:, v, 0] = dot(embed[SOS], W_proj[v]) + b_proj[v], broadcast over batch.
__global__ __launch_bounds__(256)
void sos_logits_kernel(const float* __restrict__ embed, const float* __restrict__ W_proj,
                       const float* __restrict__ b_proj, float* __restrict__ out) {
    const int v = blockIdx.x * blockDim.y + threadIdx.y;
    if (v >= VV) return;
    const int lane = threadIdx.x;
    const float* wr = W_proj + (long long)v * HB;
    float s = 0.0f;
    for (int k = lane; k < HB; k += 32) s += embed[k] * wr[k];
    for (int off = 16; off; off >>= 1) s += __shfl_down(s, off, 32);
    s = __shfl(s, 0, 32);
    const float val = s + b_proj[v];
    out[(long long)lane * (VV * TT) + (long long)v * TT] = val;
    out[(long long)(lane + 32) * (VV * TT) + (long long)v * TT] = val;
}

// ---------------------------------------------------------------------------
extern "C" void kernel_launch(void* const* d_in, const int* in_sizes, int n_in,
                              void* d_out, int out_size, void* d_ws, size_t ws_size,
                              hipStream_t stream) {
    const float* img    = (const float*)d_in[0];
    const float* pooled = (const float*)d_in[1];
    const float* embed  = (const float*)d_in[2];
    const float* W_ih   = (const float*)d_in[3];
    const float* W_hh   = (const float*)d_in[4];
    const float* b_ih   = (const float*)d_in[5];
    const float* b_hh   = (const float*)d_in[6];
    const float* W_proj = (const float*)d_in[7];
    const float* b_proj = (const float*)d_in[8];
    const float* W_e2d  = (const float*)d_in[9];
    const float* b_e2d  = (const float*)d_in[10];
    const float* W_q    = (const float*)d_in[11];
    const float* b_q    = (const float*)d_in[12];
    const float* W_k    = (const float*)d_in[13];
    const float* b_k    = (const float*)d_in[14];
    const float* W_v    = (const float*)d_in[15];
    const float* b_v    = (const float*)d_in[16];
    const float* W_h    = (const float*)d_in[17];
    const float* b_h    = (const float*)d_in[18];
    const float* W_c    = (const float*)d_in[19];
    const float* b_c    = (const float*)d_in[20];
    float* out = (float*)d_out;

    float* ws = (float*)d_ws;
    size_t off = 0;
    auto alloc = [&](size_t n) { float* p = ws + off; off += n; return p; };
    float* keys   = alloc((size_t)BB * HB * HB);   // 64 MB
    float* values = alloc((size_t)BB * HB * HB);   // 64 MB
    float* hbuf   = alloc(BB * HB);
    float* cbuf   = alloc(BB * HB);
    float* hc     = alloc(2 * BB * HB);            // [hL ; cL] contiguous (M=128)
    float* hL     = hc;
    float* cL     = hc + BB * HB;
    float* xbuf   = alloc(BB * HB);
    float* gates  = alloc((size_t)BB * 4 * HB);
    float* logits = alloc((size_t)BB * VV);
    float* qbuf   = alloc(2 * BB * HB);            // [q_h ; q_c]
    float* ctxbuf = alloc(2 * BB * HB);            // [ctx_h ; ctx_c]
    float* ctxh   = ctxbuf;
    float* ctxc   = ctxbuf + BB * HB;

    const dim3 blk(32, 8);
    const long long OUT_RS = (long long)VV * TT;
    const long long OUT_CS = TT;
    const size_t LDS1 = (size_t)64 * (HB + APAD) * 4;        // one 64-row A source
    const size_t LDS2 = 2 * LDS1;                            // two A sources / 128 rows

    encoder_kv_kernel<<<(BB * HB) / 16, blk, 0, stream>>>(
        img, W_e2d, b_e2d, W_k, b_k, W_v, b_v, keys, values);

    init_state_kernel<<<(BB * HB + 255) / 256, 256, 0, stream>>>(pooled, embed, hbuf, cbuf, xbuf);
    sos_logits_kernel<<<(VV + 7) / 8, blk, 0, stream>>>(embed, W_proj, b_proj, out);

    for (int t = 0; t < TT - 1; ++t) {
        // gates = x @ W_ih^T + h @ W_hh^T + b_ih + b_hh   [64, 2048]
        gemm_stripe_kernel<4><<<(128 + 7) / 8, blk, LDS2, stream>>>(
            xbuf, HB, W_ih, HB, hbuf, HB, W_hh, HB,
            b_ih, b_hh, gates, 4 * HB, nullptr, 0, 0, 4 * HB, 0);

        lstm_update_kernel<<<(BB * HB + 255) / 256, 256, 0, stream>>>(gates, cbuf, hL, cL);

        // logits = hL @ W_proj^T + b_proj; also scattered as res[:, :, t+1]
        gemm_stripe_kernel<4><<<(625 + 7) / 8, blk, LDS1, stream>>>(
            hL, HB, W_proj, HB, nullptr, 0, nullptr, 0,
            b_proj, nullptr, logits, VV, out + (t + 1), OUT_RS, OUT_CS, VV, 0);

        argmax_gather_kernel<<<BB, 256, 0, stream>>>(logits, embed, xbuf);

        // q = tanh([hL;cL] @ W_q^T + b_q)  (M = 128, one launch for both attends)
        gemm_stripe_kernel<8><<<(32 + 7) / 8, blk, LDS2, stream>>>(
            hc, HB, W_q, HB, nullptr, 0, nullptr, 0,
            b_q, nullptr, qbuf, HB, nullptr, 0, 0, HB, 1);

        attn_kernel<<<dim3(BB, 2), 256, 0, stream>>>(qbuf, keys, values, ctxbuf);

        // h' = tanh([ctx_h, hL] @ W_h^T + b_h) -> hbuf
        gemm_stripe_kernel<4><<<(32 + 7) / 8, blk, LDS2, stream>>>(
            ctxh, HB, W_h, 2 * HB, hL, HB, W_h + HB, 2 * HB,
            b_h, nullptr, hbuf, HB, nullptr, 0, 0, HB, 1);

        // c' = tanh([ctx_c, cL] @ W_c^T + b_c) -> cbuf
        gemm_stripe_kernel<4><<<(32 + 7) / 8, blk, LDS2, stream>>>(
            ctxc, HB, W_c, 2 * HB, cL, HB, W_c + HB, 2 * HB,
            b_c, nullptr, cbuf, HB, nullptr, 0, 0, HB, 1);
    }
}